// HGSA_v11_86947317940651
// MI455X (gfx1250) — compile-verified
//
#include <hip/hip_runtime.h>
#include <math.h>

// ---------------- problem constants ----------------
#define BB 2
#define CC 64
#define HH 512
#define WW 512
#define NHEADS 4
#define H2 256
#define W2D 256
#define HW2 65536     // 256*256
#define HWF 262144    // 512*512

typedef __attribute__((ext_vector_type(2))) float v2f;
typedef __attribute__((ext_vector_type(8))) float v8f;

__device__ __forceinline__ v8f wmma4(v2f a, v2f b, v8f c) {
    // V_WMMA_F32_16X16X4_F32 : D = A(16x4) * B(4x16) + C(16x16)
    return __builtin_amdgcn_wmma_f32_16x16x4_f32(
        false, a, false, b, (short)0, c, false, false);
}

// ---- CDNA5 async memory->LDS copy (GLOBAL_LOAD_ASYNC_TO_LDS_B128, ASYNCcnt) ----
__device__ __forceinline__ uint32_t lds_off(const void* p) {
    // generic LDS pointer: addr[31:0] is the LDS byte offset (ISA 10.2 aperture rules)
    return (uint32_t)(uintptr_t)p;
}
__device__ __forceinline__ void async_b128(uint32_t lds_byte_off, const float* gaddr) {
    asm volatile("global_load_async_to_lds_b128 %0, %1, off"
                 :: "v"(lds_byte_off), "v"(gaddr)
                 : "memory");
}
__device__ __forceinline__ void wait_async0() {
#if __has_builtin(__builtin_amdgcn_s_wait_asynccnt)
    __builtin_amdgcn_s_wait_asynccnt(0);
#else
    asm volatile("s_wait_asynccnt 0x0" ::: "memory");
#endif
}

// ---------------------------------------------------------------------------
// Kernel 1: depthwise q conv (3x3 s2) + depthwise a conv + fused pointwise a
// One block = 4 output pixels x all 64 channels.
// ---------------------------------------------------------------------------
__global__ void dw_qa_kernel(const float* __restrict__ x,
                             const float* __restrict__ wq, const float* __restrict__ bq,
                             const float* __restrict__ wadw, const float* __restrict__ badw,
                             const float* __restrict__ wapw, const float* __restrict__ bapw,
                             float* __restrict__ qout, float* __restrict__ aout) {
    int bid = blockIdx.x;                 // B * 256 * 64 blocks
    int xt = bid & 63;
    int oy = (bid >> 6) & 255;
    int b  = bid >> 14;
    int tid = threadIdx.x;
    int c = tid & 63;
    int p = tid >> 6;                     // 0..3
    int ox = xt * 4 + p;

    __shared__ float adw_s[4][64];

    float accq = 0.f, acca = 0.f;
    #pragma unroll
    for (int ky = 0; ky < 3; ++ky) {
        int iy = 2 * oy - 1 + ky;
        if (iy < 0 || iy >= HH) continue;
        #pragma unroll
        for (int kx = 0; kx < 3; ++kx) {
            int ix = 2 * ox - 1 + kx;
            if (ix < 0 || ix >= WW) continue;
            float xv = x[((b * CC + c) * HH + iy) * WW + ix];
            accq += wq[c * 9 + ky * 3 + kx] * xv;
            acca += wadw[c * 9 + ky * 3 + kx] * xv;
        }
    }
    qout[(b * CC + c) * HW2 + oy * W2D + ox] = accq + bq[c];
    adw_s[p][c] = acca + badw[c];
    __syncthreads();

    if (c < 32) {
        float s = bapw[c];
        #pragma unroll 8
        for (int i = 0; i < 64; ++i) s += wapw[c * 64 + i] * adw_s[p][i];
        aout[(b * 32 + c) * HW2 + oy * W2D + ox] = s;
    }
}

// ---------------------------------------------------------------------------
// Kernel 2: dense 3x3 stride-2 conv (k) via implicit-GEMM f32 WMMA.
// Block = 256 thr (8 waves): M=64 out-ch x N=32 pixels tile.
// (staging stays on the regular path: the -1 halo makes rows 16B-misaligned)
// ---------------------------------------------------------------------------
__global__ void kconv_wmma_kernel(const float* __restrict__ x,
                                  const float* __restrict__ wk,
                                  const float* __restrict__ bk,
                                  float* __restrict__ kout) {
    int bid = blockIdx.x;                 // B * 256 * 8
    int xt = bid & 7;
    int oy = (bid >> 3) & 255;
    int b  = bid >> 11;
    int ox0 = xt * 32;

    __shared__ float Xs[64][68];          // Xs[c][j] = x[b,c,iy, 2*ox0-1+j], j in 0..64

    int tid = threadIdx.x;
    int wave = tid >> 5, lane = tid & 31;
    int hl = lane >> 4, l15 = lane & 15;
    int mg = wave & 3, ng = wave >> 2;

    v8f acc = {0.f,0.f,0.f,0.f,0.f,0.f,0.f,0.f};

    for (int ky = 0; ky < 3; ++ky) {
        int iy = 2 * oy - 1 + ky;
        __syncthreads();
        for (int idx = tid; idx < 64 * 68; idx += 256) {
            int c = idx / 68, j = idx % 68;
            int ix = 2 * ox0 - 1 + j;
            float v = 0.f;
            if (j < 65 && iy >= 0 && iy < HH && ix >= 0 && ix < WW)
                v = x[((b * CC + c) * HH + iy) * WW + ix];
            Xs[c][j] = v;
        }
        __syncthreads();
        for (int kx = 0; kx < 3; ++kx) {
            #pragma unroll 4
            for (int i0 = 0; i0 < 64; i0 += 4) {
                v2f av, bvec;
                int m = mg * 16 + l15;
                int ka = i0 + 2 * hl;
                av.x = wk[(m * 64 + ka)     * 9 + ky * 3 + kx];
                av.y = wk[(m * 64 + ka + 1) * 9 + ky * 3 + kx];
                int j = 2 * (ng * 16 + l15) + kx;
                bvec.x = Xs[ka][j];
                bvec.y = Xs[ka + 1][j];
                acc = wmma4(av, bvec, acc);
            }
        }
    }
    #pragma unroll
    for (int r = 0; r < 8; ++r) {
        int m = mg * 16 + r + 8 * hl;
        int n = ox0 + ng * 16 + l15;
        kout[(b * CC + m) * HW2 + oy * W2D + n] = acc[r] + bk[m];
    }
}

// ---------------------------------------------------------------------------
// Kernel 3a: zero workspace region
// ---------------------------------------------------------------------------
__global__ void zero_kernel(float* __restrict__ p, int n) {
    int i = blockIdx.x * 256 + threadIdx.x;
    if (i < n) p[i] = 0.f;
}

// ---------------------------------------------------------------------------
// Kernel 3b: Gram X*X^T per (b,h) where X = [q(16); k(16); a(8); 0(8)] x 65536.
// Tiles needed: (0,0),(1,1),(2,2) diag norms; (0,2)=q.aT; (2,1)=a.kT.
// LDS staging via GLOBAL_LOAD_ASYNC_TO_LDS_B128 (rows are 64-aligned, in-bounds).
// ---------------------------------------------------------------------------
#define GCHUNK 2048
__global__ void gram_wmma_kernel(const float* __restrict__ q,
                                 const float* __restrict__ k,
                                 const float* __restrict__ a,
                                 float* __restrict__ G) {
    const int nchunks = HW2 / GCHUNK;     // 32
    int bid = blockIdx.x;                 // B*4*32 = 256
    int nc = bid % nchunks;
    int h  = (bid / nchunks) & 3;
    int b  = bid / (nchunks * 4);
    int n0 = nc * GCHUNK;

    __shared__ float Xs[48][68];          // 272B row stride: 16B-aligned rows

    int tid = threadIdx.x;
    int wave = tid >> 5, lane = tid & 31;
    int hl = lane >> 4, l15 = lane & 15;

    const int TR[5] = {0, 1, 2, 0, 2};
    const int TC[5] = {0, 1, 2, 2, 1};

    // zero the padding rows 40..47 once (never overwritten by async loads)
    for (int i = tid; i < 8 * 64; i += 256) Xs[40 + (i >> 6)][i & 63] = 0.f;

    v8f acc = {0.f,0.f,0.f,0.f,0.f,0.f,0.f,0.f};

    for (int n = n0; n < n0 + GCHUNK; n += 64) {
        __syncthreads();                  // previous iteration's reads done
        for (int i = tid; i < 40 * 16; i += 256) {
            int row = i >> 4;
            int s4  = (i & 15) * 4;       // 16B segment within 64-float row chunk
            const float* rb;
            if (row < 16)      rb = q + (size_t)(b * CC + h * 16 + row)      * HW2;
            else if (row < 32) rb = k + (size_t)(b * CC + h * 16 + row - 16) * HW2;
            else               rb = a + (size_t)(b * 32 + h * 8  + row - 32) * HW2;
            async_b128(lds_off(&Xs[row][s4]), rb + n + s4);
        }
        wait_async0();
        __syncthreads();
        if (wave < 5) {
            int tr = TR[wave], tc = TC[wave];
            #pragma unroll 4
            for (int k0 = 0; k0 < 64; k0 += 4) {
                v2f av, bvec;
                int kk = k0 + 2 * hl;
                av.x   = Xs[tr * 16 + l15][kk];
                av.y   = Xs[tr * 16 + l15][kk + 1];
                bvec.x = Xs[tc * 16 + l15][kk];
                bvec.y = Xs[tc * 16 + l15][kk + 1];
                acc = wmma4(av, bvec, acc);
            }
        }
    }
    if (wave < 5) {
        int tr = TR[wave], tc = TC[wave];
        float* Gb = G + (b * 4 + h) * 48 * 48;
        #pragma unroll
        for (int r = 0; r < 8; ++r) {
            int m = tr * 16 + r + 8 * hl;
            int nn = tc * 16 + l15;
            atomicAdd(&Gb[m * 48 + nn], acc[r]);
        }
    }
}

// ---------------------------------------------------------------------------
// Kernel 4: tiny per-(b,h) attention: normalize, softmax, W2 = SA @ SK (16x16)
// ---------------------------------------------------------------------------
__global__ void attn_small_kernel(const float* __restrict__ G,
                                  const float* __restrict__ temp_a,
                                  const float* __restrict__ temp_v,
                                  float* __restrict__ W2) {
    int bh = blockIdx.x;                  // B*4
    int h = bh & 3;
    const float* Gb = G + bh * 48 * 48;
    __shared__ float snrm[48];
    __shared__ float sSA[16][8];
    __shared__ float sSK[8][16];
    int tid = threadIdx.x;

    if (tid < 48) {
        float s = Gb[tid * 48 + tid];
        float n = sqrtf(fmaxf(s, 0.f));
        snrm[tid] = fmaxf(n, 1e-12f);
    }
    __syncthreads();

    float ta = temp_a[h], tv = temp_v[h];
    if (tid < 128) {                      // attn_a logits: [c=16][d=8]
        int c = tid >> 3, d = tid & 7;
        sSA[c][d] = Gb[c * 48 + 32 + d] / (snrm[c] * snrm[32 + d]) * ta;
    } else {                              // attn_k logits: [d=8][c=16]
        int t = tid - 128; int d = t >> 4, c = t & 15;
        sSK[d][c] = Gb[(32 + d) * 48 + 16 + c] / (snrm[32 + d] * snrm[16 + c]) * tv;
    }
    __syncthreads();

    float mine = 0.f;
    if (tid < 128) {
        int c = tid >> 3, d = tid & 7;
        float mx = sSA[c][0];
        for (int i = 1; i < 8; ++i) mx = fmaxf(mx, sSA[c][i]);
        float s = 0.f;
        for (int i = 0; i < 8; ++i) s += expf(sSA[c][i] - mx);
        mine = expf(sSA[c][d] - mx) / s;
    } else {
        int t = tid - 128; int d = t >> 4, c = t & 15;
        float mx = sSK[d][0];
        for (int i = 1; i < 16; ++i) mx = fmaxf(mx, sSK[d][i]);
        float s = 0.f;
        for (int i = 0; i < 16; ++i) s += expf(sSK[d][i] - mx);
        mine = expf(sSK[d][c] - mx) / s;
    }
    __syncthreads();
    if (tid < 128) { int c = tid >> 3, d = tid & 7; sSA[c][d] = mine; }
    else           { int t = tid - 128; sSK[t >> 4][t & 15] = mine; }
    __syncthreads();

    int c = tid >> 4, cc = tid & 15;      // 256 outputs
    float s = 0.f;
    #pragma unroll
    for (int d = 0; d < 8; ++d) s += sSA[c][d] * sSK[d][cc];
    W2[bh * 256 + c * 16 + cc] = s;
}

// ---------------------------------------------------------------------------
// Kernel 5: fold final 1x1 conv into block-diag W2: Wob[b] = wo @ blockdiag(W2)
// ---------------------------------------------------------------------------
__global__ void wob_kernel(const float* __restrict__ wo,
                           const float* __restrict__ W2,
                           float* __restrict__ Wob) {
    int idx = blockIdx.x * 256 + threadIdx.x;   // B*64*64
    if (idx >= BB * 64 * 64) return;
    int col = idx & 63;
    int r   = (idx >> 6) & 63;
    int b   = idx >> 12;
    int h = col >> 4, j = col & 15;
    const float* W2b = W2 + (b * 4 + h) * 256;
    float s = 0.f;
    #pragma unroll
    for (int i = 0; i < 16; ++i) s += wo[r * 64 + h * 16 + i] * W2b[i * 16 + j];
    Wob[idx] = s;
}

// ---------------------------------------------------------------------------
// Kernel 6: fused per-pixel double GEMM via WMMA:
//   out = Wob[b] @ ( (wv @ x + bv) * illu ) + bo      (M=64, K=64, N=32/block)
// Weight & pixel tiles staged with async-to-LDS b128 copies.
// Stride-68 weight rows kill the 16-way bank conflict on the A-operand reads.
// ---------------------------------------------------------------------------
__global__ void fused_out_wmma_kernel(const float* __restrict__ x,
                                      const float* __restrict__ illu,
                                      const float* __restrict__ wv,
                                      const float* __restrict__ bias_v,
                                      const float* __restrict__ Wob,
                                      const float* __restrict__ bo,
                                      float* __restrict__ out) {
    int bid = blockIdx.x;                 // B*512*16
    int xt = bid & 15;
    int y  = (bid >> 4) & 511;
    int b  = bid >> 13;
    int x0 = xt * 32;

    __shared__ float Xs[64][36];          // 144B rows: 16B aligned
    __shared__ float Us[64][36];
    __shared__ float Wv_s[64][68];        // 272B rows: 16B aligned + conflict-free
    __shared__ float Wo_s[64][68];

    int tid = threadIdx.x;
    int wave = tid >> 5, lane = tid & 31;
    int hl = lane >> 4, l15 = lane & 15;
    int mg = wave & 3, ng = wave >> 2;

    // async stage: wv (64x64), Wob[b] (64x64), x tile (64x32)
    for (int i = tid; i < 1024; i += 256) {
        int r = i >> 4, s4 = (i & 15) * 4;
        async_b128(lds_off(&Wv_s[r][s4]), wv + r * 64 + s4);
        async_b128(lds_off(&Wo_s[r][s4]), Wob + b * 4096 + r * 64 + s4);
    }
    for (int i = tid; i < 512; i += 256) {
        int c = i >> 3, s4 = (i & 7) * 4;
        async_b128(lds_off(&Xs[c][s4]),
                   x + (((size_t)(b * CC + c) * HH + y) * WW + x0 + s4));
    }
    wait_async0();
    __syncthreads();

    v8f acc = {0.f,0.f,0.f,0.f,0.f,0.f,0.f,0.f};
    #pragma unroll 4
    for (int k0 = 0; k0 < 64; k0 += 4) {
        v2f av, bvec;
        int m = mg * 16 + l15;
        int kk = k0 + 2 * hl;
        av.x = Wv_s[m][kk];       av.y = Wv_s[m][kk + 1];
        bvec.x = Xs[kk][ng * 16 + l15];
        bvec.y = Xs[kk + 1][ng * 16 + l15];
        acc = wmma4(av, bvec, acc);
    }
    #pragma unroll
    for (int r = 0; r < 8; ++r) {
        int m = mg * 16 + r + 8 * hl;
        int px = x0 + ng * 16 + l15;
        float u = (acc[r] + bias_v[m]) * illu[((b * CC + m) * HH + y) * WW + px];
        Us[m][ng * 16 + l15] = u;
    }
    __syncthreads();

    v8f acc2 = {0.f,0.f,0.f,0.f,0.f,0.f,0.f,0.f};
    #pragma unroll 4
    for (int k0 = 0; k0 < 64; k0 += 4) {
        v2f av, bvec;
        int m = mg * 16 + l15;
        int kk = k0 + 2 * hl;
        av.x = Wo_s[m][kk];       av.y = Wo_s[m][kk + 1];
        bvec.x = Us[kk][ng * 16 + l15];
        bvec.y = Us[kk + 1][ng * 16 + l15];
        acc2 = wmma4(av, bvec, acc2);
    }
    #pragma unroll
    for (int r = 0; r < 8; ++r) {
        int m = mg * 16 + r + 8 * hl;
        int px = x0 + ng * 16 + l15;
        out[((b * CC + m) * HH + y) * WW + px] = acc2[r] + bo[m];
    }
}

// ---------------------------------------------------------------------------
// Host launcher
// ---------------------------------------------------------------------------
extern "C" void kernel_launch(void* const* d_in, const int* in_sizes, int n_in,
                              void* d_out, int out_size, void* d_ws, size_t ws_size,
                              hipStream_t stream) {
    const float* x     = (const float*)d_in[0];
    const float* illu  = (const float*)d_in[1];
    const float* wq    = (const float*)d_in[2];
    const float* bq    = (const float*)d_in[3];
    const float* wk    = (const float*)d_in[4];
    const float* bk    = (const float*)d_in[5];
    const float* wa_dw = (const float*)d_in[6];
    const float* ba_dw = (const float*)d_in[7];
    const float* wa_pw = (const float*)d_in[8];
    const float* ba_pw = (const float*)d_in[9];
    const float* wv    = (const float*)d_in[10];
    const float* bv    = (const float*)d_in[11];
    const float* wo    = (const float*)d_in[12];
    const float* bo    = (const float*)d_in[13];
    const float* tmpa  = (const float*)d_in[14];
    const float* tmpv  = (const float*)d_in[15];
    float* out = (float*)d_out;

    float* ws = (float*)d_ws;
    // workspace layout (floats)
    const size_t q_off   = 0;                               // B*64*65536
    const size_t k_off   = q_off + (size_t)BB * 64 * HW2;   // B*64*65536
    const size_t a_off   = k_off + (size_t)BB * 64 * HW2;   // B*32*65536
    const size_t G_off   = a_off + (size_t)BB * 32 * HW2;   // B*4*48*48
    const size_t W2_off  = G_off + (size_t)BB * 4 * 48 * 48;// B*4*256
    const size_t Wob_off = W2_off + (size_t)BB * 4 * 256;   // B*64*64
    float* q_ws   = ws + q_off;
    float* k_ws   = ws + k_off;
    float* a_ws   = ws + a_off;
    float* G_ws   = ws + G_off;
    float* W2_ws  = ws + W2_off;
    float* Wob_ws = ws + Wob_off;

    // 1) depthwise q + depthwise/pointwise a
    dw_qa_kernel<<<BB * 256 * 64, 256, 0, stream>>>(
        x, wq, bq, wa_dw, ba_dw, wa_pw, ba_pw, q_ws, a_ws);

    // 2) dense 3x3 s2 conv k (WMMA implicit GEMM)
    kconv_wmma_kernel<<<BB * 256 * 8, 256, 0, stream>>>(x, wk, bk, k_ws);

    // 3) Gram accumulation (zero first; harness poisons ws each run)
    const int Gn = BB * 4 * 48 * 48;
    zero_kernel<<<(Gn + 255) / 256, 256, 0, stream>>>(G_ws, Gn);
    gram_wmma_kernel<<<BB * 4 * (HW2 / GCHUNK), 256, 0, stream>>>(q_ws, k_ws, a_ws, G_ws);

    // 4) per-(b,h) softmaxes + W2 = SA@SK
    attn_small_kernel<<<BB * 4, 256, 0, stream>>>(G_ws, tmpa, tmpv, W2_ws);

    // 5) fold wo with block-diagonal W2
    wob_kernel<<<(BB * 64 * 64 + 255) / 256, 256, 0, stream>>>(wo, W2_ws, Wob_ws);

    // 6) fused v-conv * illu then Wob GEMM + bo (WMMA x2, async-staged LDS)
    fused_out_wmma_kernel<<<BB * 512 * 16, 256, 0, stream>>>(
        x, illu, wv, bv, Wob_ws, bo, out);

    (void)in_sizes; (void)n_in; (void)out_size; (void)ws_size;
}